// GraphProjection_29850022707588
// MI455X (gfx1250) — compile-verified
//
#include <hip/hip_runtime.h>
#include <stdint.h>

// GraphProjection for MI455X (gfx1250, wave32).
// Memory-bound gather/interp kernel: one wave per point; f4 (100,352 B) staged
// into LDS per block via CDNA5 async global->LDS loads (ASYNCcnt path).
// Gathers vectorized to b128; output uses non-temporal stores (TH=NT).

#define BLOCK_THREADS   256
#define WAVES_PER_BLOCK (BLOCK_THREADS / 32)

#define F4_S      7
#define F4_C      512
#define F4_FLOATS (F4_S * F4_S * F4_C)   /* 25088 */
#define F4_BYTES  (F4_FLOATS * 4)        /* 100352 */
#define OUT_ROW   963                    /* 3 + 64 + 128 + 256 + 512 */

// Bilinear sample of one feature level for one point; lanes stripe groups of
// 4 consecutive channels (b128 gathers; corner bases are multiples of C floats
// so 16B alignment is guaranteed). Matches the reference exactly: floor/ceil
// weights (zero total weight at integer coords), JAX-style index clamp.
template <int S, int C, bool PREFETCH, typename FP>
__device__ __forceinline__ void bilerp_level(FP feat, float h, float w,
                                             float* __restrict__ outp, int lane)
{
    const float scale = (float)S * (1.0f / 224.0f);  // exact power of two
    float x = h * scale;
    float y = w * scale;
    float x1 = floorf(x), x2 = ceilf(x);
    float y1 = floorf(y), y2 = ceilf(y);
    int x1i = (int)x1;
    int x2i = (int)x2; if (x2i > S - 1) x2i = S - 1;
    int y1i = (int)y1;
    int y2i = (int)y2; if (y2i > S - 1) y2i = S - 1;
    float dx2 = x2 - x, dx1 = x - x1;
    float dy2 = y2 - y, dy1 = y - y1;
    float w11 = dx2 * dy2, w21 = dx1 * dy2, w12 = dx2 * dy1, w22 = dx1 * dy1;
    FP q11 = feat + (x1i * S + y1i) * C;
    FP q21 = feat + (x2i * S + y1i) * C;
    FP q12 = feat + (x1i * S + y2i) * C;
    FP q22 = feat + (x2i * S + y2i) * C;
    if constexpr (PREFETCH) {
        __builtin_prefetch(q11, 0, 0);
        __builtin_prefetch(q21, 0, 0);
        __builtin_prefetch(q12, 0, 0);
        __builtin_prefetch(q22, 0, 0);
    }
#pragma unroll
    for (int g = lane; g < C / 4; g += 32) {
        const int c = g * 4;
        float4 a = *(const float4*)(q11 + c);   // global_load_b128 / ds_load_b128
        float4 b = *(const float4*)(q21 + c);
        float4 e = *(const float4*)(q12 + c);
        float4 d = *(const float4*)(q22 + c);
        // Output rows are 963 floats -> level offsets not 16B aligned:
        // scalar NT stores (4 per lane-group still cover contiguous 512B/wave).
        __builtin_nontemporal_store(w11 * a.x + w21 * b.x + w12 * e.x + w22 * d.x, outp + c + 0);
        __builtin_nontemporal_store(w11 * a.y + w21 * b.y + w12 * e.y + w22 * d.y, outp + c + 1);
        __builtin_nontemporal_store(w11 * a.z + w21 * b.z + w12 * e.z + w22 * d.z, outp + c + 2);
        __builtin_nontemporal_store(w11 * a.w + w21 * b.w + w12 * e.w + w22 * d.w, outp + c + 3);
    }
}

__global__ void __launch_bounds__(BLOCK_THREADS)
graph_projection_kernel(const float* __restrict__ coord,
                        const float* __restrict__ f1,
                        const float* __restrict__ f2,
                        const float* __restrict__ f3,
                        const float* __restrict__ f4,
                        float* __restrict__ out, int N)
{
    extern __shared__ float lds_f4[];   // F4_BYTES dynamic LDS

    // ---- Stage f4 (7x7x512 fp32) into LDS ----
#if defined(__HIP_DEVICE_COMPILE__) && __has_builtin(__builtin_amdgcn_global_load_async_to_lds_b128)
    {
        // Builtin expects (int4 AS1*, int4 AS3*, imm offset, imm cpol).
        typedef int v4i __attribute__((vector_size(16)));
        __attribute__((address_space(1))) v4i* gsrc =
            (__attribute__((address_space(1))) v4i*)(v4i*)(char*)f4;
        __attribute__((address_space(3))) v4i* ldst =
            (__attribute__((address_space(3))) v4i*)(v4i*)lds_f4;
        for (int i = threadIdx.x; i < F4_FLOATS / 4; i += BLOCK_THREADS) {
            __builtin_amdgcn_global_load_async_to_lds_b128(gsrc + i, ldst + i,
                                                           /*offset=*/0, /*cpol=*/0);
        }
#if __has_builtin(__builtin_amdgcn_s_wait_asynccnt)
        __builtin_amdgcn_s_wait_asynccnt(0);
#else
        asm volatile("s_wait_asynccnt 0x0" ::: "memory");
#endif
    }
#else
    // Fallback staging path (also used by the host-side parse).
    for (int i = threadIdx.x; i < F4_FLOATS / 4; i += BLOCK_THREADS) {
        ((float4*)lds_f4)[i] = ((const float4*)f4)[i];
    }
#endif
    __syncthreads();

    const int lane = threadIdx.x & 31;
    const int wid  = threadIdx.x >> 5;
    const int p    = blockIdx.x * WAVES_PER_BLOCK + wid;
    if (p >= N) return;

    const float X = coord[3 * p + 0];
    const float Y = coord[3 * p + 1];
    const float Z = coord[3 * p + 2];

    // h = 112 * -Y / -Z + 111.5 ; w = 112 * X / -Z + 111.5 ; clip to [0,223]
    const float nZ = -Z;
    float h = 112.0f * (-Y) / nZ + 111.5f;
    float w = 112.0f * X / nZ + 111.5f;
    h = fminf(fmaxf(h, 0.0f), 223.0f);
    w = fminf(fmaxf(w, 0.0f), 223.0f);

    float* outrow = out + (size_t)p * OUT_ROW;
    if (lane == 0) {
        __builtin_nontemporal_store(X, outrow + 0);
        __builtin_nontemporal_store(Y, outrow + 1);
        __builtin_nontemporal_store(Z, outrow + 2);
    }

    bilerp_level<56, 64,  true >(f1, h, w, outrow + 3,   lane);   // L2-resident
    bilerp_level<28, 128, true >(f2, h, w, outrow + 67,  lane);   // L2-resident
    bilerp_level<14, 256, true >(f3, h, w, outrow + 195, lane);   // L2-resident
    bilerp_level<F4_S, F4_C, false>((const float*)lds_f4, h, w, outrow + 451, lane); // LDS
}

extern "C" void kernel_launch(void* const* d_in, const int* in_sizes, int n_in,
                              void* d_out, int out_size, void* d_ws, size_t ws_size,
                              hipStream_t stream)
{
    (void)n_in; (void)out_size; (void)d_ws; (void)ws_size;
    const float* coord = (const float*)d_in[0];
    const float* f1    = (const float*)d_in[1];
    const float* f2    = (const float*)d_in[2];
    const float* f3    = (const float*)d_in[3];
    const float* f4    = (const float*)d_in[4];
    float* out = (float*)d_out;

    const int N = in_sizes[0] / 3;                       // 200000 points
    const int blocks = (N + WAVES_PER_BLOCK - 1) / WAVES_PER_BLOCK;
    graph_projection_kernel<<<blocks, BLOCK_THREADS, F4_BYTES, stream>>>(
        coord, f1, f2, f3, f4, out, N);
}